// GNN_35716948034358
// MI455X (gfx1250) — compile-verified
//
#include <hip/hip_runtime.h>

#define N_NODES   100000
#define IN_DIM    128
#define HID       64
#define N_CLASSES 40
#define K_LAYERS  3

typedef __attribute__((ext_vector_type(2))) float v2f;
typedef __attribute__((ext_vector_type(8))) float v8f;

// D = A(16x4 f32) x B(4x16 f32) + C(16x16 f32), wave32 WMMA
__device__ __forceinline__ v8f wmma_f32_k4(v2f a, v2f b, v8f c) {
    // 8 args: (neg_a, A, neg_b, B, c_mod, C, reuse_a, reuse_b)
    return __builtin_amdgcn_wmma_f32_16x16x4_f32(
        false, a, false, b, (short)0, c, false, false);
}

// out[M x NCOLS] = act( (A (+ A2)) [M x KDIM] @ W [KDIM x NCOLS] + bias )
// Block = 8 waves; each wave computes one 16-row strip across all NTN n-tiles.
// W is staged in LDS, zero-padded to NTN*16 cols, and K-pair interleaved:
//   sW[(k/2)*(2*NW) + c*2 + (k&1)] = W[k][c]
// so each B fragment (W[ka][c], W[ka+1][c]) is one aligned ds_load_b64.
template<int KDIM, int NCOLS, int NTN, bool HAS_A2, bool RELU>
__global__ __launch_bounds__(256) void gnn_wmma_gemm(
    const float* __restrict__ A, const float* __restrict__ A2,
    const float* __restrict__ W, const float* __restrict__ bias,
    float* __restrict__ out, int M)
{
    constexpr int NW    = NTN * 16;              // padded column count
    constexpr int PITCH = 2 * NW;                // dwords per K-pair row
    __shared__ float sW[KDIM * NW];

    // cooperative staging: e = kpair*PITCH + c*2 + j  ->  W[kpair*2 + j][c]
    for (int e = threadIdx.x; e < KDIM * NW; e += 256) {
        const int kpair = e / PITCH;
        const int rem   = e % PITCH;
        const int c     = rem >> 1;
        const int k     = (kpair << 1) | (rem & 1);
        sW[e] = (c < NCOLS) ? W[k * NCOLS + c] : 0.0f;
    }
    __syncthreads();

    const int waveId = threadIdx.x >> 5;
    const int lane   = threadIdx.x & 31;
    const int mt     = blockIdx.x * 8 + waveId;
    if (mt >= (M >> 4)) return;                  // wave-uniform: EXEC stays all-ones

    const int mBase = mt << 4;
    const int m     = mBase + (lane & 15);       // A fragment row
    const int cLoc  = lane & 15;                 // B/C fragment column-in-tile
    const int kSel  = (lane >> 4) << 1;          // lanes 0-15 -> K{0,1}; 16-31 -> K{2,3}

    v8f acc[NTN];
#pragma unroll
    for (int t = 0; t < NTN; ++t) acc[t] = (v8f){};

    const float* __restrict__ Arow  = A + (size_t)m * KDIM;
    const float* __restrict__ A2row = HAS_A2 ? (A2 + (size_t)m * KDIM) : A;

    // base LDS dword index for this lane: kpair = kSel/2 = (lane>>4)
    const int sBase = (lane >> 4) * PITCH + cLoc * 2;

    for (int k = 0; k < KDIM; k += 4) {
        const int ka = k + kSel;
        v2f a;
        a.x = Arow[ka];
        a.y = Arow[ka + 1];
        if (HAS_A2) {
            a.x += A2row[ka];
            a.y += A2row[ka + 1];
        }
        const int sRow = (k >> 1) * PITCH + sBase;   // (k/2 + lane>>4) * PITCH + c*2
#pragma unroll
        for (int t = 0; t < NTN; ++t) {
            const v2f b = *(const v2f*)&sW[sRow + t * 32];
            acc[t] = wmma_f32_k4(a, b, acc[t]);
        }
    }

    const int rowHi = (lane >> 4) << 3;          // C/D: lanes 16-31 hold M = r+8
#pragma unroll
    for (int t = 0; t < NTN; ++t) {
        const int  col   = t * 16 + cLoc;
        const bool colOk = (col < NCOLS);        // compile-time true for full tiles
        const float bv   = colOk ? bias[col] : 0.0f;
#pragma unroll
        for (int r = 0; r < 8; ++r) {
            float v = acc[t][r] + bv;
            if (RELU) v = fmaxf(v, 0.0f);
            const int row = mBase + rowHi + r;
            if (colOk) out[(size_t)row * NCOLS + col] = v;
        }
    }
}

// agg[dst[e]][:] += h[src[e]][:]   one wave per edge, 2 coalesced steps of 32 feats
__global__ __launch_bounds__(256) void gnn_edge_scatter_add(
    const float* __restrict__ h, const int* __restrict__ src,
    const int* __restrict__ dst, float* __restrict__ agg, int nEdges)
{
    const int warpId = (blockIdx.x * blockDim.x + threadIdx.x) >> 5;
    const int lane   = threadIdx.x & 31;
    const int nWarps = (gridDim.x * blockDim.x) >> 5;
    for (int e = warpId; e < nEdges; e += nWarps) {
        const int s = src[e];
        const int d = dst[e];
        const float v0 = h[(size_t)s * HID + lane];
        const float v1 = h[(size_t)s * HID + lane + 32];
        atomicAdd(&agg[(size_t)d * HID + lane],      v0);
        atomicAdd(&agg[(size_t)d * HID + lane + 32], v1);
    }
}

__global__ __launch_bounds__(256) void gnn_zero_f32(float* __restrict__ p, int n)
{
    int i = blockIdx.x * blockDim.x + threadIdx.x;
    const int stride = gridDim.x * blockDim.x;
    for (; i < n; i += stride) p[i] = 0.0f;
}

extern "C" void kernel_launch(void* const* d_in, const int* in_sizes, int n_in,
                              void* d_out, int out_size, void* d_ws, size_t ws_size,
                              hipStream_t stream)
{
    (void)n_in; (void)out_size; (void)ws_size;

    const float* x        = (const float*)d_in[0];
    const int*   edge     = (const int*)  d_in[1];   // [2, E] row-major
    const float* W_in     = (const float*)d_in[2];
    const float* b_in     = (const float*)d_in[3];
    const float* W_layers = (const float*)d_in[4];   // [K, 64, 64]
    const float* b_layers = (const float*)d_in[5];   // [K, 64]
    const float* W_cls    = (const float*)d_in[6];   // [64, 40]
    const float* b_cls    = (const float*)d_in[7];   // [40]
    float*       out      = (float*)d_out;

    const int E    = in_sizes[1] / 2;
    const int* src = edge;
    const int* dst = edge + E;

    float* h0  = (float*)d_ws;
    float* h1  = h0 + (size_t)N_NODES * HID;
    float* agg = h1 + (size_t)N_NODES * HID;

    const int mtiles = N_NODES / 16;                 // 6250, exact
    const int blocks = (mtiles + 7) / 8;             // 8 waves (strips) per block

    // ---- h = relu(x @ W_in + b_in)  [100000x128 @ 128x64] ----
    gnn_wmma_gemm<IN_DIM, HID, HID / 16, false, true>
        <<<blocks, 256, 0, stream>>>(x, nullptr, W_in, b_in, h0, N_NODES);

    float* hc = h0;
    float* hn = h1;
    for (int l = 0; l < K_LAYERS; ++l) {
        gnn_zero_f32<<<4096, 256, 0, stream>>>(agg, N_NODES * HID);
        gnn_edge_scatter_add<<<2048, 256, 0, stream>>>(hc, src, dst, agg, E);
        gnn_wmma_gemm<HID, HID, HID / 16, true, true>
            <<<blocks, 256, 0, stream>>>(hc, agg,
                                         W_layers + (size_t)l * HID * HID,
                                         b_layers + (size_t)l * HID, hn, N_NODES);
        float* t = hc; hc = hn; hn = t;
    }

    // ---- out = h @ W_cls + b_cls  [100000x64 @ 64x40], 3 ragged n-tiles ----
    gnn_wmma_gemm<HID, N_CLASSES, 3, false, false>
        <<<blocks, 256, 0, stream>>>(hc, nullptr, W_cls, b_cls, out, N_NODES);
}